// CR8_reg_2stage_13975823582044
// MI455X (gfx1250) — compile-verified
//
#include <hip/hip_runtime.h>
#include <hip/hip_bf16.h>

#define H_DIM 192
#define W_DIM 256
#define HW    49152   // 192*256

typedef float v2f __attribute__((ext_vector_type(2)));
typedef float v8f __attribute__((ext_vector_type(8)));

__device__ __forceinline__ float lrelu(float x) { return x > 0.0f ? x : 0.01f * x; }

__device__ __forceinline__ v8f wmma4(v2f a, v2f b, v8f c) {
    // D = A(16x4 f32) * B(4x16 f32) + C(16x16 f32)
    return __builtin_amdgcn_wmma_f32_16x16x4_f32(false, a, false, b, (short)0, c,
                                                 false, false);
}

// B-operand loads from pair-interleaved weights WP[k/2][o][2]:
// lane needs rows (kb, kb+1) at column `col`, kb = 4*kk + 2*hl -> one b64 load.
template<int KCH>
__device__ __forceinline__ void loadB(const float* __restrict__ WP, int O, int col,
                                      int hl, v2f* b)
{
    const float* p = WP + (size_t)col * 2 + (size_t)hl * 2 * O;
#pragma unroll
    for (int kk = 0; kk < KCH; ++kk)
        b[kk] = *(const v2f*)(p + (size_t)kk * 4 * O);
}

// One GEMM pass: A (KCH*4 x 16 pixels) in regs, B double-buffered across the
// N-tiles (load next tile's B while current WMMA chain runs), D (+bias,
// optional lrelu) written to a 16x132-padded LDS tile [pixel][chan].
// Bias values are preloaded up front so no late scalar load forces a full
// loadcnt drain inside the pipelined loop.
template<int KCH, int NTILES, bool RELU>
__device__ __forceinline__ void gemm_store(const v2f* a, const float* __restrict__ WP,
                                           int O, const float* __restrict__ bias,
                                           float* ldsT, int lane)
{
    const int hl = lane >> 4;      // half-wave
    const int nn = lane & 15;

    // preload all per-tile bias values BEFORE the B pipeline starts
    float bvals[NTILES];
#pragma unroll
    for (int nt = 0; nt < NTILES; ++nt)
        bvals[nt] = bias[nt * 16 + nn];

    v2f bb[2][KCH];
    loadB<KCH>(WP, O, nn, hl, bb[0]);

#pragma unroll
    for (int nt = 0; nt < NTILES; ++nt) {
        if (nt + 1 < NTILES)
            loadB<KCH>(WP, O, (nt + 1) * 16 + nn, hl, bb[(nt + 1) & 1]);
        const v2f* cur = bb[nt & 1];

        const int n0 = nt * 16;
        v8f acc;
#pragma unroll
        for (int i = 0; i < 8; ++i) acc[i] = bvals[nt];
#pragma unroll
        for (int kk = 0; kk < KCH; ++kk)
            acc = wmma4(a[kk], cur[kk], acc);
#pragma unroll
        for (int i = 0; i < 8; ++i) {
            float v = acc[i];
            if (RELU) v = lrelu(v);
            // D layout: VGPR i -> (M=i, lanes 0-15) and (M=i+8, lanes 16-31)
            ldsT[(i + hl * 8) * 132 + n0 + nn] = v;
        }
    }
}

// Load A-operand regs from the LDS tile [pixel][chan]:
// lanes 0-15: (M=lane, K=4k+0/1), lanes 16-31: (M=lane-16, K=4k+2/3)
template<int KCH>
__device__ __forceinline__ void load_a(const float* ldsT, v2f* a, int lane)
{
    const int hl = lane >> 4;
    const int m  = lane & 15;
    const float* row = ldsT + m * 132 + 2 * hl;
#pragma unroll
    for (int kk = 0; kk < KCH; ++kk)
        a[kk] = *(const v2f*)(row + 4 * kk);
}

// Load A-operand regs for the x_in tile (channel-major [128][HW])
__device__ __forceinline__ void load_a_xin(const float* __restrict__ x_in, int p0,
                                           int lane, v2f* a)
{
    const int hl = lane >> 4;
    const int m  = lane & 15;
    const float* xp = x_in + p0 + m;
#pragma unroll
    for (int kk = 0; kk < 32; ++kk) {
        const int kb = 4 * kk + 2 * hl;
        a[kk].x = xp[(size_t)kb * HW];
        a[kk].y = xp[(size_t)(kb + 1) * HW];
    }
}

// Copy 16x128 LDS tile to workspace rows at n = w*H + h (reference's x_l order)
__device__ __forceinline__ void copy_rows(const float* ldsT, float* __restrict__ dst,
                                          int p0, int lane)
{
#pragma unroll
    for (int mm = 0; mm < 16; ++mm) {
        const int p  = p0 + mm;
        const int hh = p >> 8;        // p / 256
        const int ww = p & 255;       // p % 256
        const size_t n = (size_t)ww * H_DIM + hh;
        float4 v = *(const float4*)(ldsT + mm * 132 + lane * 4);
        *(float4*)(dst + n * 128 + (size_t)lane * 4) = v;
    }
}

// [O][K] -> pair-interleaved [K/2][O][2] so B-operand pairs are one b64 load
__global__ void kTransposePair(const float* __restrict__ src, float* __restrict__ dst,
                               int O, int K)
{
    int i = blockIdx.x * blockDim.x + threadIdx.x;
    if (i < O * K) {
        int o = i / K;
        int k = i - o * K;
        dst[(size_t)(k >> 1) * 2 * O + o * 2 + (k & 1)] = src[i];
    }
}

// ---------------- Stage 1: all dense GEMMs (WMMA), one wave per 16 pixels ----
__global__ __launch_bounds__(128) void fused_stage1(
    const float* __restrict__ x_in,
    const float* __restrict__ WPcl1,  const float* __restrict__ bcl1,
    const float* __restrict__ WPcl21, const float* __restrict__ bcl21,
    const float* __restrict__ WPcl31, const float* __restrict__ bcl31,
    const float* __restrict__ WPr1,   const float* __restrict__ br1,
    const float* __restrict__ WPm1,   const float* __restrict__ bm1,
    const float* __restrict__ WPm2,   const float* __restrict__ bm2,
    const float* __restrict__ Wm3,    const float* __restrict__ bm3,
    float* __restrict__ x1_ws, float* __restrict__ rl_ws,
    int* __restrict__ inds1_ws, float* __restrict__ mask_out)
{
    __shared__ float lds[4 * 16 * 132];
    const int lane = threadIdx.x & 31;
    const int wid  = threadIdx.x >> 5;
    const int tile = blockIdx.x * 4 + wid;
    const int p0   = tile * 16;                 // pixels p0..p0+15, p = h*W + w
    float* ldsT = lds + wid * 16 * 132;

    v2f a[32];

    // x1 = lrelu(Wcl1 @ x)  (128 -> 128)
    load_a_xin(x_in, p0, lane, a);
    gemm_store<32, 8, true>(a, WPcl1, 128, bcl1, ldsT, lane);
    __syncthreads();
    copy_rows(ldsT, x1_ws, p0, lane);
    load_a<32>(ldsT, a, lane);
    __syncthreads();

    // x2 = lrelu(Wcl21 @ x1)
    gemm_store<32, 8, true>(a, WPcl21, 128, bcl21, ldsT, lane);
    __syncthreads();
    load_a<32>(ldsT, a, lane);
    __syncthreads();

    // cl1 = Wcl31 @ x2 + b  (128 -> 32, no relu)
    gemm_store<32, 2, false>(a, WPcl31, 32, bcl31, ldsT, lane);
    __syncthreads();

    // inds1 = argmax over 32 channels (first-max tie-break), stored in p-order
    if (lane < 16) {
        const float* row = ldsT + lane * 132;
        float best = row[0];
        int bi = 0;
#pragma unroll
        for (int c = 1; c < 32; ++c) {
            float v = row[c];
            if (v > best) { best = v; bi = c; }
        }
        inds1_ws[p0 + lane] = bi;
    }
    __syncthreads();

    // r = lrelu(Wr1 @ x)  (128 -> 128); reload x_in A (L2-hot) to cap VGPRs
    load_a_xin(x_in, p0, lane, a);
    gemm_store<32, 8, true>(a, WPr1, 128, br1, ldsT, lane);
    __syncthreads();
    copy_rows(ldsT, rl_ws, p0, lane);
    __syncthreads();

    // mask branch: m1 = lrelu(Wm1 @ x)  (128 -> 32); `a` still holds x_in
    gemm_store<32, 2, true>(a, WPm1, 32, bm1, ldsT, lane);
    __syncthreads();
    v2f am[8];
    load_a<8>(ldsT, am, lane);
    __syncthreads();

    // m2 = lrelu(Wm2 @ m1)  (32 -> 16)
    gemm_store<8, 1, true>(am, WPm2, 16, bm2, ldsT, lane);
    __syncthreads();

    // mask = lrelu(Wm3 @ m2 + b)  (16 -> 1), output at h*W+w
    if (lane < 16) {
        const float* row = ldsT + lane * 132;
        float s = bm3[0];
#pragma unroll
        for (int c = 0; c < 16; ++c) s += row[c] * Wm3[c];
        mask_out[p0 + lane] = lrelu(s);
    }
}

// ---------------- Stage 2: data-dependent cond_mul chain, one wave per row ---
__global__ __launch_bounds__(256) void fused_stage2(
    const float* __restrict__ x1_ws, const float* __restrict__ rl_ws,
    const int* __restrict__ inds1_ws,
    const float* __restrict__ Wcl22, const float* __restrict__ bcl22,
    const float* __restrict__ Wcl32, const float* __restrict__ bcl32,
    const float* __restrict__ Wr2,   const float* __restrict__ br2,
    const float* __restrict__ Wr3,   const float* __restrict__ br3,
    float* __restrict__ out)
{
    __shared__ float lds[8 * 128];
    const int lane = threadIdx.x & 31;
    const int wid  = threadIdx.x >> 5;
    const int n    = blockIdx.x * 8 + wid;
    float* lx = lds + wid * 128;

    int idx1 = inds1_ws[n];
    idx1 = __builtin_amdgcn_readfirstlane(idx1);

    // stage x_l row into LDS
    {
        float4 v = *(const float4*)(x1_ws + (size_t)n * 128 + lane * 4);
        *(float4*)(lx + lane * 4) = v;
    }
    __syncthreads();

    // h = lrelu(x_l @ Wcl22[idx1] + b)  (lane = output channel)
    const float* Wp = Wcl22 + (size_t)idx1 * 128 * 32;
    float acc = bcl22[idx1 * 32 + lane];
#pragma unroll 4
    for (int c = 0; c < 128; ++c)
        acc = fmaf(lx[c], Wp[c * 32 + lane], acc);
    float h = lrelu(acc);

    // h2 = h @ Wcl32[idx1] + b
    const float* W2p = Wcl32 + (size_t)idx1 * 32 * 32;
    float acc2 = bcl32[idx1 * 32 + lane];
#pragma unroll
    for (int c = 0; c < 32; ++c)
        acc2 = fmaf(__shfl(h, c, 32), W2p[c * 32 + lane], acc2);

    // argmax over lanes (first index on tie, like jnp.argmax)
    float bv = acc2;
    int   bi = lane;
#pragma unroll
    for (int off = 16; off >= 1; off >>= 1) {
        float ov = __shfl_xor(bv, off, 32);
        int   oi = __shfl_xor(bi, off, 32);
        if (ov > bv || (ov == bv && oi < bi)) { bv = ov; bi = oi; }
    }
    const int inds   = idx1 * 32 + bi;
    const int inds_r = min(max(inds, 0), 1023);
    const int isup   = min(inds_r >> 7, 7);   // CLASS_FACTOR == 128

    __syncthreads();
    // stage r_l row into LDS (reuse buffer)
    {
        float4 v = *(const float4*)(rl_ws + (size_t)n * 128 + lane * 4);
        *(float4*)(lx + lane * 4) = v;
    }
    __syncthreads();

    // r2 = lrelu(r_l @ Wr2[isup] + b)
    const float* Wrp = Wr2 + (size_t)isup * 128 * 32;
    float accr = br2[isup * 32 + lane];
#pragma unroll 4
    for (int c = 0; c < 128; ++c)
        accr = fmaf(lx[c], Wrp[c * 32 + lane], accr);
    float r2 = lrelu(accr);

    // regression = r2 @ Wr3[inds_r] + b  (32 -> 1, wave reduction)
    float t = r2 * Wr3[inds_r * 32 + lane];
#pragma unroll
    for (int off = 16; off >= 1; off >>= 1)
        t += __shfl_xor(t, off, 32);
    const float reg = t + br3[inds_r];
    const float xr  = ((float)inds + reg) * (1.0f / 1024.0f);
    if (lane == 0) out[n] = xr;
}

extern "C" void kernel_launch(void* const* d_in, const int* in_sizes, int n_in,
                              void* d_out, int out_size, void* d_ws, size_t ws_size,
                              hipStream_t stream)
{
    (void)in_sizes; (void)n_in; (void)out_size; (void)ws_size;

    const float* x_in  = (const float*)d_in[0];
    const float* Wm1   = (const float*)d_in[1];
    const float* bm1   = (const float*)d_in[2];
    const float* Wm2   = (const float*)d_in[3];
    const float* bm2   = (const float*)d_in[4];
    const float* Wm3   = (const float*)d_in[5];
    const float* bm3   = (const float*)d_in[6];
    const float* Wcl1  = (const float*)d_in[7];
    const float* bcl1  = (const float*)d_in[8];
    const float* Wcl21 = (const float*)d_in[9];
    const float* bcl21 = (const float*)d_in[10];
    const float* Wcl31 = (const float*)d_in[11];
    const float* bcl31 = (const float*)d_in[12];
    const float* Wcl22 = (const float*)d_in[13];
    const float* bcl22 = (const float*)d_in[14];
    const float* Wcl32 = (const float*)d_in[15];
    const float* bcl32 = (const float*)d_in[16];
    const float* Wr1   = (const float*)d_in[17];
    const float* br1   = (const float*)d_in[18];
    const float* Wr2   = (const float*)d_in[19];
    const float* br2   = (const float*)d_in[20];
    const float* Wr3   = (const float*)d_in[21];
    const float* br3   = (const float*)d_in[22];

    float* wsf = (float*)d_ws;
    size_t off = 0;
    float* x1_ws  = wsf + off; off += (size_t)HW * 128;   // x_l rows (n = w*H+h)
    float* rl_ws  = wsf + off; off += (size_t)HW * 128;   // r_l rows
    int*   i1_ws  = (int*)(wsf + off); off += HW;         // inds1 (p-order)
    float* wpCL1  = wsf + off; off += 128 * 128;
    float* wpCL21 = wsf + off; off += 128 * 128;
    float* wpR1   = wsf + off; off += 128 * 128;
    float* wpCL31 = wsf + off; off += 128 * 32;
    float* wpM1   = wsf + off; off += 128 * 32;
    float* wpM2   = wsf + off; off += 32 * 16;

    // [O][K] -> pair-interleaved [K/2][O][2] for b64 B-operand loads
    kTransposePair<<<64, 256, 0, stream>>>(Wcl1,  wpCL1, 128, 128);
    kTransposePair<<<64, 256, 0, stream>>>(Wcl21, wpCL21, 128, 128);
    kTransposePair<<<64, 256, 0, stream>>>(Wr1,   wpR1, 128, 128);
    kTransposePair<<<16, 256, 0, stream>>>(Wcl31, wpCL31, 32, 128);
    kTransposePair<<<16, 256, 0, stream>>>(Wm1,   wpM1, 32, 128);
    kTransposePair<<<2,  256, 0, stream>>>(Wm2,   wpM2, 16, 32);

    float* out = (float*)d_out;           // [0,HW): x_real ; [HW,2*HW): mask

    fused_stage1<<<HW / 16 / 4, 128, 0, stream>>>(
        x_in, wpCL1, bcl1, wpCL21, bcl21, wpCL31, bcl31,
        wpR1, br1, wpM1, bm1, wpM2, bm2, Wm3, bm3,
        x1_ws, rl_ws, i1_ws, out + HW);

    fused_stage2<<<HW / 8, 256, 0, stream>>>(
        x1_ws, rl_ws, i1_ws, Wcl22, bcl22, Wcl32, bcl32,
        Wr2, br2, Wr3, br3, out);
}